// WeightedBoxesFusionProccessor_15461882266077
// MI455X (gfx1250) — compile-verified
//
#include <hip/hip_runtime.h>
#include <cstdint>

#define PRE        2000
#define POST       300
#define NSORT      2048   // next pow2 >= PRE
#define BLOCK      1024
#define IOU_THRESH 0.55f

// Single-workgroup sequential WBF scan, all state LDS-resident.
// LDS budget: 48000 (x) + 32000 (sum_wb f4) + 8000 (sum_s) + 8000 (cnt|cls)
//           + 8192 (keys) + 8192 (idx) + 8 = 112392 B  (< gfx1250 LDS limit)
__global__ __launch_bounds__(BLOCK)
void wbf_kernel(const float* __restrict__ x, float* __restrict__ out) {
  __shared__ float    s_x[PRE * 6];       // staged detections (async DMA target)
  __shared__ float4   s_sumwb[PRE];       // weighted box sums (16B -> ds_*_b128)
  __shared__ float    s_sums[PRE];        // score sums
  __shared__ unsigned s_cntcls[PRE];      // [15:0]=count, [31:16]=class
  __shared__ float    s_key[NSORT];       // sort keys (fused score, -1 invalid)
  __shared__ int      s_idx[NSORT];       // sort payload (cluster index)
  __shared__ int      s_comm[2];          // [0]=min matched idx, [1]=n_valid

  const int tid = threadIdx.x;

  // ---- Stage input into LDS with CDNA5 async global->LDS DMA ----
  {
    const uint64_t gbase = (uint64_t)(const void*)x;
    const unsigned lbase = (unsigned)(uint64_t)(const void*)&s_x[0]; // LDS offset
    for (int k = tid; k < (PRE * 6) / 4; k += BLOCK) {       // 3000 x b128
      unsigned laddr = lbase + (unsigned)k * 16u;
      uint64_t gaddr = gbase + (uint64_t)k * 16u;
      asm volatile("global_load_async_to_lds_b128 %0, %1, off"
                   :: "v"(laddr), "v"(gaddr) : "memory");
    }
  }

  // Zero cluster state while the async DMA is in flight.
  for (int j = tid; j < PRE; j += BLOCK) {
    s_sumwb[j]  = make_float4(0.0f, 0.0f, 0.0f, 0.0f);
    s_sums[j]   = 0.0f;
    s_cntcls[j] = 0u;
  }
  if (tid == 0) { s_comm[0] = 0x7FFFFFFF; s_comm[1] = 0; }

#if __has_builtin(__builtin_amdgcn_s_wait_asynccnt)
  __builtin_amdgcn_s_wait_asynccnt(0);
#else
  asm volatile("s_wait_asynccnt 0" ::: "memory");
#endif
  __syncthreads();

  // ---- Sequential scan over detections; lane-parallel cluster matching ----
  for (int i = 0; i < PRE; ++i) {
    const float bx1 = s_x[i * 6 + 0];
    const float by1 = s_x[i * 6 + 1];
    const float bx2 = s_x[i * 6 + 2];
    const float by2 = s_x[i * 6 + 3];
    const float sc  = s_x[i * 6 + 4];
    const int   cli = (int)s_x[i * 6 + 5];
    const float a1  = (bx2 - bx1) * (by2 - by1);

    const int nv = s_comm[1];

    for (int j = tid; j < nv; j += BLOCK) {   // valid clusters form a prefix
      unsigned vc = s_cntcls[j];
      if ((int)(vc >> 16) == cli) {
        float4 w  = s_sumwb[j];               // single ds_load_b128
        float ss  = s_sums[j];
        float inv = 1.0f / ss;                // ss > 0 for valid clusters
        float m0 = w.x * inv;
        float m1 = w.y * inv;
        float m2 = w.z * inv;
        float m3 = w.w * inv;
        float iw = fminf(bx2, m2) - fmaxf(bx1, m0);
        float ih = fminf(by2, m3) - fmaxf(by1, m1);
        iw = fmaxf(iw, 0.0f); ih = fmaxf(ih, 0.0f);
        float inter = iw * ih;
        float a2    = (m2 - m0) * (m3 - m1);
        float uni   = a1 + a2 - inter;
        float iou   = (uni > 0.0f) ? (inter / uni) : 0.0f;
        if (iou > IOU_THRESH) atomicMin(&s_comm[0], j);  // ds_min: "first" match
      }
    }
    __syncthreads();

    if (tid == 0) {
      int idx = s_comm[0];
      bool found = (idx != 0x7FFFFFFF);
      if (!found) idx = nv;                    // append new cluster
      float4 w = s_sumwb[idx];
      w.x += sc * bx1;
      w.y += sc * by1;
      w.z += sc * bx2;
      w.w += sc * by2;
      s_sumwb[idx] = w;
      s_sums[idx] += sc;
      unsigned vc = s_cntcls[idx];
      s_cntcls[idx] = (((vc & 0xFFFFu) + 1u) & 0xFFFFu) | ((unsigned)cli << 16);
      if (!found) s_comm[1] = nv + 1;
      s_comm[0] = 0x7FFFFFFF;
    }
    __syncthreads();
  }

  // ---- Build sort keys: fused score for valid clusters, -1 otherwise ----
  {
    const int nv = s_comm[1];
    for (int k = tid; k < NSORT; k += BLOCK) {
      float key = -1.0f;
      if (k < nv) {
        unsigned vc = s_cntcls[k];
        float cnt = (float)(vc & 0xFFFFu);
        key = s_sums[k] / fmaxf(cnt, 1.0f);
      }
      s_key[k] = key;
      s_idx[k] = k;
    }
  }
  __syncthreads();

  // ---- Bitonic sort, descending (2048 elems, 1024 threads) ----
  for (unsigned k2 = 2; k2 <= NSORT; k2 <<= 1) {
    for (unsigned jj = k2 >> 1; jj > 0; jj >>= 1) {
      for (unsigned l = (unsigned)tid; l < NSORT; l += BLOCK) {
        unsigned p = l ^ jj;
        if (p > l) {
          float a = s_key[l], b = s_key[p];
          bool desc = ((l & k2) == 0);
          if (desc ? (a < b) : (a > b)) {
            s_key[l] = b; s_key[p] = a;
            int t = s_idx[l]; s_idx[l] = s_idx[p]; s_idx[p] = t;
          }
        }
      }
      __syncthreads();
    }
  }

  // ---- Emit top-300 rows: [box(4), score, class], zeros where invalid ----
  for (int r = tid; r < POST; r += BLOCK) {
    int   j   = s_idx[r];
    float key = s_key[r];
    float o0 = 0.0f, o1 = 0.0f, o2 = 0.0f, o3 = 0.0f, o4 = 0.0f, o5 = 0.0f;
    if (key >= 0.0f) {                        // valid scores are > 0
      float4 w  = s_sumwb[j];                 // single ds_load_b128
      float inv = 1.0f / s_sums[j];
      o0 = w.x * inv;
      o1 = w.y * inv;
      o2 = w.z * inv;
      o3 = w.w * inv;
      o4 = key;
      o5 = (float)(s_cntcls[j] >> 16);
    }
    out[r * 6 + 0] = o0;
    out[r * 6 + 1] = o1;
    out[r * 6 + 2] = o2;
    out[r * 6 + 3] = o3;
    out[r * 6 + 4] = o4;
    out[r * 6 + 5] = o5;
  }
}

extern "C" void kernel_launch(void* const* d_in, const int* in_sizes, int n_in,
                              void* d_out, int out_size, void* d_ws, size_t ws_size,
                              hipStream_t stream) {
  (void)in_sizes; (void)n_in; (void)out_size; (void)d_ws; (void)ws_size;
  const float* x   = (const float*)d_in[0];
  float*       out = (float*)d_out;
  wbf_kernel<<<dim3(1), dim3(BLOCK), 0, stream>>>(x, out);
}